// Sparsity_49984829391100
// MI455X (gfx1250) — compile-verified
//
#include <hip/hip_runtime.h>
#include <stdint.h>

// 2:4 structured-sparsity masking, fp32, HBM-bound (~537 MB traffic -> ~23us floor).
// CDNA5 path: double-buffered GLOBAL_LOAD_ASYNC_TO_LDS_B128 (ASYNCcnt) pipeline,
// ds_load_b128 readback, lane-local majority-vote top-2 select, NT b128 stores.

typedef float v4f __attribute__((ext_vector_type(4)));

#define THREADS 256
#define CHUNKS  4
#define PER_BLOCK (THREADS * CHUNKS)

// Keep top-2 of 4 by |magnitude|; ties keep the lower index (matches jax.lax.top_k).
// Element i is zeroed iff a majority (>=2) of its 3 competitors beat it, where
// j beats i iff a_j > a_i, or a_j == a_i && j < i  (reverse comparisons are pure
// negations, so only 6 v_cmp are needed; majority folds to scalar mask logic).
__device__ __forceinline__ v4f top2_of_4(v4f v) {
  float a0 = __builtin_fabsf(v.x), a1 = __builtin_fabsf(v.y),
        a2 = __builtin_fabsf(v.z), a3 = __builtin_fabsf(v.w);
  bool c01 = a1 > a0, c02 = a2 > a0, c03 = a3 > a0;
  bool c12 = a2 > a1, c13 = a3 > a1, c23 = a3 > a2;
  bool z0 = (c01 & c02) | (c01 & c03) | (c02 & c03);
  bool z1 = ((!c01) & c12) | ((!c01) & c13) | (c12 & c13);
  bool z2 = ((!c02) & (!c12)) | ((!c02) & c23) | ((!c12) & c23);
  bool z3 = ((!c03) & (!c13)) | ((!c03) & (!c23)) | ((!c13) & (!c23));
  v4f r;
  r.x = z0 ? 0.0f : v.x;
  r.y = z1 ? 0.0f : v.y;
  r.z = z2 ? 0.0f : v.z;
  r.w = z3 ? 0.0f : v.w;
  return r;
}

__global__ __launch_bounds__(THREADS) void nm24_sparsity_async_kernel(
    const v4f* __restrict__ in4, v4f* __restrict__ out4, unsigned nblk4) {
  __shared__ v4f stage[2][THREADS];      // 2 x 4KB double buffer
  const unsigned tid = threadIdx.x;
  const unsigned blockBase = blockIdx.x * PER_BLOCK;

  if (blockBase + PER_BLOCK <= nblk4) {
    // Full tile: async double-buffered pipeline, no bounds checks.
    const unsigned base = blockBase + tid;
    unsigned lds_a0 = (unsigned)(uintptr_t)&stage[0][tid];  // low 32b of generic
    unsigned lds_a1 = (unsigned)(uintptr_t)&stage[1][tid];  // ptr == LDS offset

    {
      uint64_t g = (uint64_t)(uintptr_t)(in4 + base);
      asm volatile("global_load_async_to_lds_b128 %0, %1, off th:TH_LOAD_NT"
                   :: "v"(lds_a0), "v"(g) : "memory");
    }
#pragma unroll
    for (int k = 0; k < CHUNKS; ++k) {
      unsigned i = base + (unsigned)k * THREADS;
      if (k + 1 < CHUNKS) {
        // Prefetch next chunk into the other buffer, then wait for current.
        uint64_t g = (uint64_t)(uintptr_t)(in4 + i + THREADS);
        unsigned lds_n = ((k + 1) & 1) ? lds_a1 : lds_a0;
        asm volatile("global_load_async_to_lds_b128 %0, %1, off th:TH_LOAD_NT"
                     :: "v"(lds_n), "v"(g) : "memory");
        asm volatile("s_wait_asynccnt 0x1" ::: "memory");
      } else {
        asm volatile("s_wait_asynccnt 0x0" ::: "memory");
      }
      v4f v = stage[k & 1][tid];                       // ds_load_b128 (own 16B)
      __builtin_nontemporal_store(top2_of_4(v), &out4[i]);
    }
  } else {
    // Tail fallback (never taken at 8192^2): guarded direct NT loads.
#pragma unroll
    for (int k = 0; k < CHUNKS; ++k) {
      unsigned i = blockBase + (unsigned)k * THREADS + tid;
      if (i < nblk4) {
        v4f v = __builtin_nontemporal_load(&in4[i]);
        __builtin_nontemporal_store(top2_of_4(v), &out4[i]);
      }
    }
  }
}

extern "C" void kernel_launch(void* const* d_in, const int* in_sizes, int n_in,
                              void* d_out, int out_size, void* d_ws, size_t ws_size,
                              hipStream_t stream) {
  const float* in = (const float*)d_in[0];   // 8192*8192 fp32
  float* out = (float*)d_out;
  (void)n_in; (void)d_ws; (void)ws_size; (void)out_size;

  unsigned n = (unsigned)in_sizes[0];        // 67,108,864 elements
  unsigned nblk4 = n / 4u;                   // 16,777,216 float4 groups
  unsigned grid = (nblk4 + PER_BLOCK - 1u) / PER_BLOCK;  // 16384 full tiles

  nm24_sparsity_async_kernel<<<grid, THREADS, 0, stream>>>(
      (const v4f*)in, (v4f*)out, nblk4);
}